// smpl_model_refined_90460601188973
// MI455X (gfx1250) — compile-verified
//
#include <hip/hip_runtime.h>
#include <math.h>

// ---------------- problem constants ----------------
#define N_V   6890
#define N_PAD 6912          // 432 tiles of 16
#define J_N   24
#define NB_N  10
#define B_N   256
#define K_DIM 96            // J*4
#define C_DIM 768           // B*3

#define M_TILES (N_PAD / 16)   // 432
#define N_TILES (C_DIM / 16)   // 48

// output layout (floats): verts | J_transformed | v_plane
#define VERTS_SIZE (B_N * N_V * 3)          // 5,291,520
#define JT_OFF     VERTS_SIZE
#define PLANE_OFF  (VERTS_SIZE + B_N * J_N * 3)

// workspace layout (floats)
#define OFF_VS 0                       // v_shaped: 6890*3 (pad 20736)
#define OFF_WP 20736                   // W': N_PAD * 96
#define OFF_JS (20736 + N_PAD * K_DIM) // J_shaped: 72 (pad 128)
#define OFF_AP (OFF_JS + 128)          // A': 96 * 768
// total = 758,144 floats ≈ 3.03 MB of d_ws

typedef __attribute__((ext_vector_type(2))) float v2f;
typedef __attribute__((ext_vector_type(8))) float v8f;

__device__ const int c_parents[J_N] =
    {-1,0,0,0,1,2,3,4,5,6,7,8,9,9,9,12,13,14,16,17,18,19,20,21};

// ---------------- kernel 1: v_shaped + W' = w[v,j]*h[v,n] ----------------
__global__ void k_shape(const float* __restrict__ vt, const float* __restrict__ sd,
                        const float* __restrict__ lbs, const float* __restrict__ betas,
                        float* __restrict__ ws) {
  int v = blockIdx.x * blockDim.x + threadIdx.x;
  if (v >= N_PAD) return;
  float* vs = ws + OFF_VS;
  float* Wp = ws + OFF_WP;
  if (v < N_V) {
    float scale = betas[0];
    float h0, h1, h2;
    float h[3];
    #pragma unroll
    for (int k = 0; k < 3; ++k) {
      float acc = vt[v * 3 + k];
      #pragma unroll
      for (int l = 0; l < NB_N; ++l) acc += betas[1 + l] * sd[v * 30 + k * 10 + l];
      h[k] = acc * scale;
      vs[v * 3 + k] = h[k];
    }
    h0 = h[0]; h1 = h[1]; h2 = h[2];
    #pragma unroll
    for (int j = 0; j < J_N; ++j) {
      float w = lbs[v * J_N + j];
      float4 r; r.x = w * h0; r.y = w * h1; r.z = w * h2; r.w = w;
      *(float4*)(Wp + (size_t)v * K_DIM + j * 4) = r;
    }
  } else {
    float4 z = {0.f, 0.f, 0.f, 0.f};
    #pragma unroll
    for (int j = 0; j < J_N; ++j)
      *(float4*)(Wp + (size_t)v * K_DIM + j * 4) = z;
  }
}

// ---------------- kernel 2a: J_shaped = J_regressor @ v_shaped ----------------
__global__ void k_jreg(const float* __restrict__ Jr, float* __restrict__ ws) {
  __shared__ float sm[256];
  int jk = blockIdx.x;              // 0..71
  int j = jk / 3, k = jk % 3;
  const float* vs = ws + OFF_VS;
  float acc = 0.f;
  for (int v = threadIdx.x; v < N_V; v += 256)
    acc += Jr[j * N_V + v] * vs[v * 3 + k];
  sm[threadIdx.x] = acc;
  __syncthreads();
  for (int s = 128; s > 0; s >>= 1) {
    if (threadIdx.x < s) sm[threadIdx.x] += sm[threadIdx.x + s];
    __syncthreads();
  }
  if (threadIdx.x == 0) ws[OFF_JS + jk] = sm[0];
}

// ------- kernel 2b: pose assembly + Rodrigues + chain -> A' and J_transformed -------
__global__ void k_chain(const float* __restrict__ bp,  const float* __restrict__ rbk,
                        const float* __restrict__ rfr, const float* __restrict__ lbk,
                        const float* __restrict__ lfr, const float* __restrict__ go,
                        float* __restrict__ ws, float* __restrict__ out) {
  int b = threadIdx.x;              // one thread per batch, 256 threads
  const float* Js = ws + OFF_JS;
  float* Ap = ws + OFF_AP;

  float fp[72];
  fp[0] = go[b * 3]; fp[1] = go[b * 3 + 1]; fp[2] = go[b * 3 + 2];
  for (int i = 0; i < 18; ++i) fp[3 + i]  = bp[b * 57 + i];
  for (int i = 0; i < 3;  ++i) fp[21 + i] = lbk[b * 3 + i];
  for (int i = 0; i < 3;  ++i) fp[24 + i] = rbk[b * 3 + i];
  for (int i = 0; i < 3;  ++i) fp[27 + i] = bp[b * 57 + 18 + i];
  fp[30] = lfr[b]; fp[31] = 0.f; fp[32] = 0.f;
  fp[33] = rfr[b]; fp[34] = 0.f; fp[35] = 0.f;
  for (int i = 0; i < 36; ++i) fp[36 + i] = bp[b * 57 + 21 + i];

  float Rc[J_N][9], tc[J_N][3];
  for (int j = 0; j < J_N; ++j) {
    float rx = fp[3 * j], ry = fp[3 * j + 1], rz = fp[3 * j + 2];
    float ax = rx + 1e-8f, ay = ry + 1e-8f, az = rz + 1e-8f;
    float ang = sqrtf(ax * ax + ay * ay + az * az);
    float inv = 1.f / ang;
    float kx = rx * inv, ky = ry * inv, kz = rz * inv;
    float s = sinf(ang), c = cosf(ang), oc = 1.f - c;
    float K[9]  = {0.f, -kz, ky,  kz, 0.f, -kx,  -ky, kx, 0.f};
    float K2[9];
    for (int m = 0; m < 3; ++m)
      for (int n = 0; n < 3; ++n) {
        float a = 0.f;
        for (int q = 0; q < 3; ++q) a += K[m * 3 + q] * K[q * 3 + n];
        K2[m * 3 + n] = a;
      }
    float Rl[9];
    for (int i = 0; i < 9; ++i)
      Rl[i] = ((i % 4 == 0) ? 1.f : 0.f) + s * K[i] + oc * K2[i];

    int par = c_parents[j];
    float rel[3];
    for (int m = 0; m < 3; ++m)
      rel[m] = Js[j * 3 + m] - ((j > 0) ? Js[par * 3 + m] : 0.f);

    if (j == 0) {
      for (int i = 0; i < 9; ++i) Rc[0][i] = Rl[i];
      for (int m = 0; m < 3; ++m) tc[0][m] = rel[m];
    } else {
      for (int m = 0; m < 3; ++m) {
        for (int n = 0; n < 3; ++n) {
          float a = 0.f;
          for (int q = 0; q < 3; ++q) a += Rc[par][m * 3 + q] * Rl[q * 3 + n];
          Rc[j][m * 3 + n] = a;
        }
        float t = tc[par][m];
        for (int q = 0; q < 3; ++q) t += Rc[par][m * 3 + q] * rel[q];
        tc[j][m] = t;
      }
    }
  }

  // A' rows (j*4+n) x cols (b*3+m); J_transformed output
  for (int j = 0; j < J_N; ++j) {
    float ta[3];
    for (int m = 0; m < 3; ++m) {
      float ib = 0.f;
      for (int n = 0; n < 3; ++n) ib += Rc[j][m * 3 + n] * Js[j * 3 + n];
      ta[m] = tc[j][m] - ib;
      out[JT_OFF + b * (J_N * 3) + j * 3 + m] = tc[j][m];
    }
    for (int n = 0; n < 4; ++n)
      for (int m = 0; m < 3; ++m)
        Ap[(j * 4 + n) * C_DIM + b * 3 + m] = (n < 3) ? Rc[j][m * 3 + n] : ta[m];
  }
}

// ---------------- kernel 3: verts GEMM via V_WMMA_F32_16X16X4_F32 ----------------
// out[v, c] = sum_k W'[v,k] * A'[k,c] ;  c = b*3+m ; + transl fused
__global__ void k_lbs_wmma(const float* __restrict__ transl,
                           const float* __restrict__ ws,
                           float* __restrict__ out) {
  const float* Wp = ws + OFF_WP;
  const float* Ap = ws + OFF_AP;

  int wave = (blockIdx.x * blockDim.x + threadIdx.x) >> 5;  // 1 wave = 1 tile
  int lane = threadIdx.x & 31;
  int mt = wave % M_TILES;
  int nt = wave / M_TILES;          // consecutive waves share nt -> B-tile reuse in L2
  int row  = lane & 15;
  int half = lane >> 4;

  // ISA A 16x4 layout: lane = M, VGPR pair selects K = half*2 + {0,1}
  const float* aB = Wp + ((size_t)(mt * 16 + row)) * K_DIM + half * 2;
  // ISA B 4x16 layout: lane = N, rows K = half*2 + {0,1}
  const float* bB = Ap + (size_t)(half * 2) * C_DIM + nt * 16 + row;

  v8f acc = {};
  #pragma unroll
  for (int s = 0; s < K_DIM / 4; ++s) {          // 24 WMMA steps
    v2f a = { aB[s * 4], aB[s * 4 + 1] };
    v2f b = { bB[(size_t)(s * 4) * C_DIM], bB[(size_t)(s * 4 + 1) * C_DIM] };
    acc = __builtin_amdgcn_wmma_f32_16x16x4_f32(
        /*neg_a=*/false, a, /*neg_b=*/false, b,
        /*c_mod=*/(short)0, acc, /*reuse_a=*/false, /*reuse_b=*/false);
  }

  int c = nt * 16 + row;            // flat (b*3 + m) == flat transl index
  float tr = transl[c];
  int bidx = c / 3, m = c - bidx * 3;
  #pragma unroll
  for (int i = 0; i < 8; ++i) {     // D: VGPR i holds M = i + 8*half
    int v = mt * 16 + half * 8 + i;
    if (v < N_V)
      out[((size_t)bidx * N_V + v) * 3 + m] = acc[i] + tr;
  }
}

// ---------------- kernel 4: plane points (gathered LBS) ----------------
__global__ void k_planes(const float* __restrict__ lbs, const float* __restrict__ transl,
                         const float* __restrict__ ipbl, const float* __restrict__ ipfl,
                         const float* __restrict__ ipbr, const float* __restrict__ ipfr,
                         const int* __restrict__ ibl, const int* __restrict__ ibr,
                         const int* __restrict__ ifl, const int* __restrict__ ifr,
                         const float* __restrict__ ws, float* __restrict__ out) {
  int t = blockIdx.x * blockDim.x + threadIdx.x;
  if (t >= B_N * 400) return;
  int b = t / 400, q = t - b * 400;
  int g = q / 100, p = q - g * 100;

  const int* ids; const float* ip;
  if      (g == 0) { ids = ibl; ip = ipbl; }
  else if (g == 1) { ids = ifl; ip = ipfl; }
  else if (g == 2) { ids = ibr; ip = ipbr; }
  else             { ids = ifr; ip = ipfr; }

  int v = ids[p];
  const float* Ap = ws + OFF_AP;

  float T[12];
  #pragma unroll
  for (int i = 0; i < 12; ++i) T[i] = 0.f;
  for (int j = 0; j < J_N; ++j) {
    float w = lbs[v * J_N + j];
    #pragma unroll
    for (int n = 0; n < 4; ++n)
      #pragma unroll
      for (int m = 0; m < 3; ++m)
        T[m * 4 + n] += w * Ap[(j * 4 + n) * C_DIM + b * 3 + m];
  }
  float ix = ip[(b * 100 + p) * 3 + 0];
  float iy = ip[(b * 100 + p) * 3 + 1];
  float iz = ip[(b * 100 + p) * 3 + 2];
  #pragma unroll
  for (int m = 0; m < 3; ++m) {
    float o = T[m * 4] * ix + T[m * 4 + 1] * iy + T[m * 4 + 2] * iz
            + T[m * 4 + 3] + transl[b * 3 + m];
    out[PLANE_OFF + (size_t)t * 3 + m] = o;
  }
}

// ---------------- launch ----------------
extern "C" void kernel_launch(void* const* d_in, const int* in_sizes, int n_in,
                              void* d_out, int out_size, void* d_ws, size_t ws_size,
                              hipStream_t stream) {
  const float* vt    = (const float*)d_in[0];
  const float* sd    = (const float*)d_in[1];
  const float* Jr    = (const float*)d_in[2];
  const float* lbs   = (const float*)d_in[3];
  const float* betas = (const float*)d_in[4];
  const float* bp    = (const float*)d_in[5];
  const float* rbk   = (const float*)d_in[6];
  const float* rfr   = (const float*)d_in[7];
  const float* lbk   = (const float*)d_in[8];
  const float* lfr   = (const float*)d_in[9];
  const float* trl   = (const float*)d_in[10];
  const float* go    = (const float*)d_in[11];
  const float* ipbl  = (const float*)d_in[12];
  const float* ipfl  = (const float*)d_in[13];
  const float* ipbr  = (const float*)d_in[14];
  const float* ipfr  = (const float*)d_in[15];
  const int*   ibl   = (const int*)d_in[17];
  const int*   ibr   = (const int*)d_in[18];
  const int*   ifl   = (const int*)d_in[19];
  const int*   ifr   = (const int*)d_in[20];

  float* ws  = (float*)d_ws;   // needs ~3.03 MB
  float* out = (float*)d_out;

  k_shape<<<(N_PAD + 255) / 256, 256, 0, stream>>>(vt, sd, lbs, betas, ws);
  k_jreg<<<72, 256, 0, stream>>>(Jr, ws);
  k_chain<<<1, 256, 0, stream>>>(bp, rbk, rfr, lbk, lfr, go, ws, out);
  k_lbs_wmma<<<(M_TILES * N_TILES) / 8, 256, 0, stream>>>(trl, ws, out);
  k_planes<<<(B_N * 400 + 255) / 256, 256, 0, stream>>>(
      lbs, trl, ipbl, ipfl, ipbr, ipfr, ibl, ibr, ifl, ifr, ws, out);
}